// Gate_16501264351574
// MI455X (gfx1250) — compile-verified
//
#include <hip/hip_runtime.h>
#include <hip/hip_bf16.h>
#include <math.h>

typedef __attribute__((ext_vector_type(2))) float v2f;
typedef __attribute__((ext_vector_type(8))) float v8f;

#define Cc   256
#define Hh   64
#define Ww   64
#define HWc  4096          // H*W
#define WEIGHTS_N 524288   // 16*8*64*64
#define COUNTS_OFF 1048576 // 2*WEIGHTS_N

__global__ void moe_gate_zero_counts(float* __restrict__ out) {
    if (threadIdx.x < 64) out[COUNTS_OFF + threadIdx.x] = 0.0f;
}

__global__ __launch_bounds__(256) void moe_gate_kernel(
    const float* __restrict__ x,     // [16,256,64,64]
    const float* __restrict__ gw,    // [64,256,3,3]
    const float* __restrict__ bias,  // [64]
    float* __restrict__ out)         // weights | indices | counts
{
    // 64KB weight-pair staging (reused as logit buffer) + 64 floats of bias
    __shared__ __align__(16) float lds[16384 + 64];
    float2* wpair = (float2*)lds;
    float*  sbias = lds + 16384;

    const int tid  = threadIdx.x;
    const int wave = tid >> 5;
    const int lane = tid & 31;
    const int lm   = lane & 15;   // pixel-in-tile / expert lane
    const int lh   = lane >> 4;   // K half-select (0 or 1)

    // 256 blocks: b in [0,16), 4-row group in [0,16)
    const int g  = blockIdx.x;
    const int b  = g >> 4;
    const int h0 = (g & 15) * 4;
    const int h_out = h0 + (wave >> 1);   // 8 waves -> 4 rows, 2 waves per row
    const int w0    = (wave & 1) * 32;    // each wave: 32 pixels = 2 M-tiles

    if (tid < 64) sbias[tid] = bias[tid];

    v8f acc[2][4];
    #pragma unroll
    for (int t = 0; t < 2; ++t)
        #pragma unroll
        for (int n = 0; n < 4; ++n)
            acc[t][n] = (v8f){};

    const float* ximg = x + (size_t)b * Cc * HWc;

    for (int kh = 0; kh < 3; ++kh) {
        for (int kw = 0; kw < 3; ++kw) {
            const int tap = kh * 3 + kw;
            __syncthreads();  // previous tap's LDS reads done
            // Stage: wpair[cp][e] = {gw[e][2cp][tap], gw[e][2cp+1][tap]}
            {
                const int e   = tid & 63;
                const int cpb = (tid >> 6) * 32;
                const float* gwe = gw + (size_t)e * (Cc * 9) + tap;
                #pragma unroll 4
                for (int j = 0; j < 32; ++j) {
                    const int cp = cpb + j;
                    float2 v;
                    v.x = gwe[(2 * cp) * 9];
                    v.y = gwe[(2 * cp + 1) * 9];
                    wpair[cp * 64 + e] = v;
                }
            }
            __syncthreads();

            const int h_in  = h_out + kh - 1;
            const bool vh   = (h_in >= 0) & (h_in < Hh);
            const int w_in0 = w0 + lm + kw - 1;
            const int w_in1 = w_in0 + 16;
            const bool v0 = vh & (w_in0 >= 0) & (w_in0 < Ww);
            const bool v1 = vh & (w_in1 < Ww);   // w_in1 >= 15 always
            // Clamped (always in-bounds) base addresses; mask after load.
            const float* xp0 = ximg + (v0 ? (h_in * Ww + w_in0) : 0);
            const float* xp1 = ximg + (v1 ? (h_in * Ww + w_in1) : 0);
            const int cl = 2 * lh;  // lanes 16-31 take K=c+2,c+3

            #pragma unroll 4
            for (int c = 0; c < Cc; c += 4) {
                const float a0x = xp0[(c + cl) * HWc];
                const float a0y = xp0[(c + cl + 1) * HWc];
                const float a1x = xp1[(c + cl) * HWc];
                const float a1y = xp1[(c + cl + 1) * HWc];
                v2f a[2];
                a[0].x = v0 ? a0x : 0.0f;
                a[0].y = v0 ? a0y : 0.0f;
                a[1].x = v1 ? a1x : 0.0f;
                a[1].y = v1 ? a1y : 0.0f;

                const float2* wp = wpair + ((c >> 1) + lh) * 64 + lm;
                v2f f[4];
                #pragma unroll
                for (int n = 0; n < 4; ++n) {
                    const float2 bb = wp[16 * n];
                    f[n].x = bb.x; f[n].y = bb.y;
                }
                #pragma unroll
                for (int t = 0; t < 2; ++t)
                    #pragma unroll
                    for (int n = 0; n < 4; ++n)
                        acc[t][n] = __builtin_amdgcn_wmma_f32_16x16x4_f32(
                            false, a[t], false, f[n], (short)0, acc[t][n], false, false);
            }
        }
    }

    // Spill logits to LDS (reuse weight buffer: 8 waves * 32 px * 64 e = 16384 floats)
    __syncthreads();
    {
        float* lg = lds + wave * 2048;
        #pragma unroll
        for (int t = 0; t < 2; ++t)
            #pragma unroll
            for (int n = 0; n < 4; ++n)
                #pragma unroll
                for (int r = 0; r < 8; ++r) {
                    const int m = t * 16 + r + 8 * lh;  // C/D layout: lanes>=16 hold M=r+8
                    lg[m * 64 + n * 16 + lm] = acc[t][n][r];
                }
    }
    __syncthreads();

    // One thread per pixel (256 px / 256 threads):
    // sigmoid -> biased top-8 -> softmax of unbiased -> stores
    {
        const int p  = tid;
        const int wv = p >> 5;
        const int m2 = p & 31;
        const int h  = h0 + (wv >> 1);
        const int w  = (wv & 1) * 32 + m2;
        float* sc = lds + wv * 2048 + m2 * 64;

        for (int e = 0; e < 64; ++e)
            sc[e] = 1.0f / (1.0f + __expf(-sc[e]));

        unsigned long long used = 0ull;
        float wvk[8]; int id[8];
        #pragma unroll
        for (int k = 0; k < 8; ++k) {
            float best = -3.0e38f; int bi = 0;
            for (int e = 0; e < 64; ++e) {
                if (!((used >> e) & 1ull)) {
                    const float v = sc[e] + sbias[e];
                    if (v > best) { best = v; bi = e; }  // first index wins ties (lax.top_k)
                }
            }
            used |= 1ull << bi;
            id[k]  = bi;
            wvk[k] = sc[bi];   // unbiased score
        }
        float mx = wvk[0];
        #pragma unroll
        for (int k = 1; k < 8; ++k) mx = fmaxf(mx, wvk[k]);
        float sum = 0.0f;
        #pragma unroll
        for (int k = 0; k < 8; ++k) { wvk[k] = __expf(wvk[k] - mx); sum += wvk[k]; }
        const float inv = 1.0f / sum;  // ROUTE_SCALE == 1.0
        #pragma unroll
        for (int k = 0; k < 8; ++k) {
            const size_t o = (((size_t)b * 8 + k) * 64 + h) * 64 + w;
            out[o]             = wvk[k] * inv;      // weights [B,K,H,W]
            out[WEIGHTS_N + o] = (float)id[k];      // indices [B,K,H,W]
            atomicAdd(&out[COUNTS_OFF + id[k]], 1.0f);  // equal addends -> deterministic
        }
    }
}

extern "C" void kernel_launch(void* const* d_in, const int* in_sizes, int n_in,
                              void* d_out, int out_size, void* d_ws, size_t ws_size,
                              hipStream_t stream) {
    (void)in_sizes; (void)n_in; (void)d_ws; (void)ws_size; (void)out_size;
    const float* x    = (const float*)d_in[0];
    const float* gw   = (const float*)d_in[1];
    const float* bias = (const float*)d_in[2];
    float* out = (float*)d_out;

    moe_gate_zero_counts<<<1, 64, 0, stream>>>(out);
    // 16 images * 16 four-row groups = 256 workgroups, 256 threads (8 wave32)
    moe_gate_kernel<<<256, 256, 0, stream>>>(x, gw, bias, out);
}